// TextEncoder_55671366091616
// MI455X (gfx1250) — compile-verified
//
#include <hip/hip_runtime.h>
#include <hip/hip_bf16.h>

// Problem constants (match reference)
#define B_   32
#define S_   2048
#define D_   1024
#define E_   64
#define K_   8
#define T_   128   // N_TAGS

// GEMM tiling: KC=32 K-chunk, double buffered in LDS.
// Row strides padded to 36 floats = 144 B: keeps every row 16-B aligned for
// async b128 LDS writes AND gives conflict-free banks for the b64 WMMA feeds
// (36m mod 64 distinct for m=0..15; hi-lane +2 offset lands on disjoint banks).
#define KC   32
#define SAP  36
#define SBP  36

typedef __attribute__((ext_vector_type(2))) float v2f;
typedef __attribute__((ext_vector_type(8))) float v8f;

// --- async global->LDS path (gfx1250), guarded so compile never breaks ------
#if defined(__AMDGCN__) && __has_builtin(__builtin_amdgcn_global_load_async_to_lds_b128)
#define HAVE_ASYNC_LDS 1
#else
#define HAVE_ASYNC_LDS 0
#endif

typedef int v4i __attribute__((vector_size(16)));                 // int4
typedef __attribute__((address_space(1))) v4i gv4i;               // global int4
typedef __attribute__((address_space(3))) v4i lv4i;               // LDS int4

__device__ __forceinline__ void cp_async16(float* ldsDst, const float* gSrc)
{
#if HAVE_ASYNC_LDS
    __builtin_amdgcn_global_load_async_to_lds_b128(
        (gv4i*)(gSrc), (lv4i*)(ldsDst), /*offset=*/0, /*cpol=*/0);
#else
    const float4 v = *(const float4*)gSrc;
    ldsDst[0] = v.x; ldsDst[1] = v.y; ldsDst[2] = v.z; ldsDst[3] = v.w;
#endif
}

__device__ __forceinline__ void wait_async_lds()
{
#if HAVE_ASYNC_LDS
#if __has_builtin(__builtin_amdgcn_s_wait_asynccnt)
    __builtin_amdgcn_s_wait_asynccnt(0);
#else
    asm volatile("s_wait_asynccnt 0x0" ::: "memory");
#endif
#endif
}

// ---------------------------------------------------------------------------
// Kernel 1: gather + masked mean-pool  ->  ent_emb [B*E, D] (f32, in d_ws)
// One 256-thread block per entity row; each thread owns one float4 column.
// HBM-bound: <= 64 MB of gathered reads, fully coalesced (16 lanes = 256 B).
// ---------------------------------------------------------------------------
__global__ void __launch_bounds__(256)
pool_kernel(const float* __restrict__ hidden,
            const int*   __restrict__ sidx,
            const int*   __restrict__ scnt,
            const int*   __restrict__ nent,
            float*       __restrict__ ent)
{
    const int r = blockIdx.x;            // 0 .. B*E-1
    const int b = r >> 6;                // /E_
    const int e = r & (E_ - 1);
    const int cnt = scnt[r];             // 1..K  (reference guarantees >=1)
    const int ne  = nent[b];
    const int t   = threadIdx.x;         // 0..255 -> float4 column

    const float4* __restrict__ hid4 = (const float4*)hidden;
    const int* __restrict__ idxp = sidx + r * K_;

    float4 acc; acc.x = 0.f; acc.y = 0.f; acc.z = 0.f; acc.w = 0.f;
#pragma unroll
    for (int k = 0; k < K_; ++k) {
        if (k < cnt) {                   // uniform per block
            const int s = idxp[k];
            const float4 v = hid4[((size_t)b * S_ + s) * (D_ / 4) + t];
            acc.x += v.x; acc.y += v.y; acc.z += v.z; acc.w += v.w;
        }
    }
    const float scale = (e < ne) ? (1.0f / (float)cnt) : 0.0f;
    float4 o; o.x = acc.x * scale; o.y = acc.y * scale;
    o.z = acc.z * scale; o.w = acc.w * scale;
    ((float4*)ent)[(size_t)r * (D_ / 4) + t] = o;
}

// ---------------------------------------------------------------------------
// Kernel 2: logits = ent_emb @ fc_w^T + fc_b  via V_WMMA_F32_16X16X4_F32
// Grid: (B*E)/16 = 128 workgroups; 256 threads = 8 waves.
// WG tile: 16 entity rows x 128 tags. Each wave owns one 16x16 C tile.
// K is processed in KC=32 chunks, DOUBLE BUFFERED: chunk i+1 is staged into
// LDS with global_load_async_to_lds_b128 (ASYNCcnt) while 8 v_wmma run on
// chunk i; then s_wait_asynccnt 0 + barrier, swap.
// LDS: 2 x (A 16x36 + B 128x36) f32  ~= 41 KB.
// ---------------------------------------------------------------------------
__global__ void __launch_bounds__(256)
gemm_kernel(const float* __restrict__ ent,
            const float* __restrict__ fc_w,
            const float* __restrict__ fc_b,
            float*       __restrict__ out)
{
    __shared__ float sA[2][16 * SAP];
    __shared__ float sB[2][T_ * SBP];

    const int tid  = threadIdx.x;
    const int wave = tid >> 5;           // 0..7 -> tag tile
    const int lane = tid & 31;
    const int r0   = blockIdx.x * 16;    // entity-row base
    const int hi   = (lane >= 16) ? 2 : 0;   // K sub-offset per ISA A/B layout
    const int mrow = lane & 15;          // A row (M) and B col (N) for this lane
    const int wtag = wave * 16;          // tag base for this wave

    v8f c = {0.f, 0.f, 0.f, 0.f, 0.f, 0.f, 0.f, 0.f};

    // ---- stage one chunk (issue copies; completion via wait_async_lds) ----
    auto stage = [&](int p, int kc) {
        // A tile: 16 rows x KC cols = 128 float4 -> threads 0..127, one each.
        if (tid < 128) {
            const int j   = tid >> 3;    // row 0..15
            const int seg = tid & 7;     // col segment (x4 floats)
            cp_async16(&sA[p][j * SAP + seg * 4],
                       ent + (size_t)(r0 + j) * D_ + kc + seg * 4);
        }
        // B tile: 128 tags x KC cols = 1024 float4 -> 4 per thread.
        for (int i = tid; i < T_ * (KC / 4); i += 256) {
            const int tag = i >> 3;
            const int seg = i & 7;
            cp_async16(&sB[p][tag * SBP + seg * 4],
                       fc_w + (size_t)tag * D_ + kc + seg * 4);
        }
    };

    int p = 0;
    stage(p, 0);
    wait_async_lds();
    __syncthreads();

    for (int kc = 0; kc < D_; kc += KC) {
        // Issue async staging of the next chunk into the other buffer while
        // the WMMAs below consume the current one.
        if (kc + KC < D_) stage(p ^ 1, kc + KC);

        const float* __restrict__ ap0 = &sA[p][mrow * SAP + hi];
        const float* __restrict__ bp0 = &sB[p][(wtag + mrow) * SBP + hi];
#pragma unroll
        for (int k0 = 0; k0 < KC; k0 += 4) {
            v2f a, bm;
            a.x  = ap0[k0];     a.y  = ap0[k0 + 1];   // A[M=mrow][k0+hi .. +1]
            bm.x = bp0[k0];     bm.y = bp0[k0 + 1];   // B[k0+hi..][N=mrow]
            c = __builtin_amdgcn_wmma_f32_16x16x4_f32(
                    /*neg_a=*/false, a, /*neg_b=*/false, bm,
                    /*c_mod=*/(short)0, c, /*reuse_a=*/false, /*reuse_b=*/false);
        }

        wait_async_lds();     // my async writes for chunk kc+KC are done
        __syncthreads();      // everyone's writes visible; reads of buf p done
        p ^= 1;
    }

    // --- epilogue: bias + store C tile ---
    const float bias = fc_b[wtag + mrow];
    const int mb = (lane < 16) ? 0 : 8;   // C layout: VGPR i -> M = i + mb
#pragma unroll
    for (int i = 0; i < 8; ++i) {
        out[(size_t)(r0 + i + mb) * T_ + wtag + mrow] = c[i] + bias;
    }
}

// ---------------------------------------------------------------------------
extern "C" void kernel_launch(void* const* d_in, const int* in_sizes, int n_in,
                              void* d_out, int out_size, void* d_ws, size_t ws_size,
                              hipStream_t stream)
{
    const float* hidden = (const float*)d_in[0];   // [B,S,D] f32
    const int*   sidx   = (const int*)  d_in[1];   // [B,E,K] i32
    const int*   scnt   = (const int*)  d_in[2];   // [B,E]   i32
    const int*   nent   = (const int*)  d_in[3];   // [B]     i32
    const float* fc_w   = (const float*)d_in[4];   // [T,D]   f32
    const float* fc_b   = (const float*)d_in[5];   // [T]     f32
    float*       out    = (float*)d_out;           // [B,E,T] f32
    float*       ent    = (float*)d_ws;            // [B*E,D] f32 scratch (8 MB)

    pool_kernel<<<dim3(B_ * E_), dim3(256), 0, stream>>>(hidden, sidx, scnt,
                                                         nent, ent);
    gemm_kernel<<<dim3((B_ * E_) / 16), dim3(256), 0, stream>>>(ent, fc_w,
                                                                fc_b, out);
}